// GATFeatureExtractor_43748536877087
// MI455X (gfx1250) — compile-verified
//
#include <hip/hip_runtime.h>
#include <hip/hip_bf16.h>
#include <math.h>

// ---------------------------------------------------------------------------
// GAT 2-layer forward for MI455X (gfx1250, wave32).
// GEMMs use V_WMMA_F32_16X16X4_F32 (exact fp32 tensor path), register-blocked
// 16x64 per wave so the A-fragment is reused across 4 WMMAs per K-step.
// Edge phase uses hardware float atomics (global_atomic_add_f32 / max trick),
// relying on the 192MB L2 to keep the 102MB feature matrix resident; per-head
// softmax coefficients are staged in LDS once per edge-block.
// ---------------------------------------------------------------------------

#define FIN    128
#define HID    64
#define HEADS  8
#define SLOPE  0.2f

typedef __attribute__((ext_vector_type(2))) float v2f;
typedef __attribute__((ext_vector_type(8))) float v8f;

// ---------------------------------------------------------------- fill ------
__global__ void fill_f32(float* __restrict__ p, float v, size_t n) {
    size_t i = (size_t)blockIdx.x * blockDim.x + threadIdx.x;
    if (i < n) p[i] = v;
}

// --------------------------------------------------------- WMMA fp32 GEMM ---
// C[M,Nc] = A[M,K] * B[K,Nc], row major. One 16x64 strip (4 tiles) per wave32.
// A-frag (16x4 f32): lane L holds A[m][k0], A[m][k0+1], m=L&15, k0=2*(L>>4).
// B-frag (4x16 f32): lane L holds B[k0][n], B[k0+1][n],   n=L&15.
// C/D (16x16 f32, 8 VGPRs): lane L, vgpr v -> row (L<16 ? v : v+8), col L&15.
// Requires Nc % 64 == 0 and K % 4 == 0 (true here: 512/64, K=128/512).
__global__ __launch_bounds__(256)
void wmma_gemm_f32(const float* __restrict__ A, const float* __restrict__ B,
                   float* __restrict__ C, int M, int K, int Nc) {
    const int wave    = threadIdx.x >> 5;
    const int lane    = threadIdx.x & 31;
    const int nstrips = Nc >> 6;                      // 64-wide strips
    const int strip   = blockIdx.x * (blockDim.x >> 5) + wave;
    const int mt      = strip / nstrips;
    const int ns      = strip - mt * nstrips;
    if (mt * 16 >= M) return;                         // wave-uniform guard

    const int n    = lane & 15;                       // A row idx == B/C col idx
    const int half = lane >> 4;                       // 0 or 1
    const float* __restrict__ arow  = A + (size_t)(mt * 16 + n) * K;
    const float* __restrict__ bbase = B + (size_t)(ns * 64 + n);

    v8f acc0 = {}, acc1 = {}, acc2 = {}, acc3 = {};
    for (int k = 0; k < K; k += 4) {
        const int ka = k + 2 * half;
        v2f a = { arow[ka], arow[ka + 1] };           // contiguous -> b64 load
        const float* __restrict__ b0 = bbase + (size_t)ka * Nc;
        const float* __restrict__ b1 = b0 + Nc;
        v2f f0 = { b0[0],  b1[0]  };
        v2f f1 = { b0[16], b1[16] };
        v2f f2 = { b0[32], b1[32] };
        v2f f3 = { b0[48], b1[48] };
        acc0 = __builtin_amdgcn_wmma_f32_16x16x4_f32(false, a, false, f0, (short)0, acc0, false, false);
        acc1 = __builtin_amdgcn_wmma_f32_16x16x4_f32(false, a, false, f1, (short)0, acc1, false, false);
        acc2 = __builtin_amdgcn_wmma_f32_16x16x4_f32(false, a, false, f2, (short)0, acc2, false, false);
        acc3 = __builtin_amdgcn_wmma_f32_16x16x4_f32(false, a, false, f3, (short)0, acc3, false, false);
    }

    float* __restrict__ crow =
        C + (size_t)(mt * 16 + half * 8) * Nc + (size_t)(ns * 64 + n);
#pragma unroll
    for (int v = 0; v < 8; ++v) {
        crow[(size_t)v * Nc +  0] = acc0[v];
        crow[(size_t)v * Nc + 16] = acc1[v];
        crow[(size_t)v * Nc + 32] = acc2[v];
        crow[(size_t)v * Nc + 48] = acc3[v];
    }
}

// ------------------------------------------------- per-node attention dots --
__global__ void node_alpha(const float* __restrict__ h,
                           const float* __restrict__ a_src,
                           const float* __restrict__ a_dst,
                           float* __restrict__ as, float* __restrict__ ad,
                           int H, int C) {
    const int n    = blockIdx.x;
    const int head = threadIdx.x >> 5;
    const int lane = threadIdx.x & 31;
    if (head >= H) return;
    const float* __restrict__ row = h + ((size_t)n * H + head) * C;
    float s = 0.f, d = 0.f;
    for (int c = lane; c < C; c += 32) {
        const float v = row[c];
        s += v * a_src[head * C + c];
        d += v * a_dst[head * C + c];
    }
#pragma unroll
    for (int off = 16; off > 0; off >>= 1) {
        s += __shfl_down(s, off, 32);
        d += __shfl_down(d, off, 32);
    }
    if (lane == 0) {
        as[(size_t)n * H + head] = s;
        ad[(size_t)n * H + head] = d;
    }
}

// --------------------------------------------------------- float atomics ----
__device__ __forceinline__ void atomicMaxF(float* addr, float val) {
    // monotonic int/uint mapping; valid with -inf initialization
    if (val >= 0.f) atomicMax((int*)addr, __float_as_int(val));
    else            atomicMin((unsigned int*)addr, __float_as_uint(val));
}

__device__ __forceinline__ float leaky(float x) {
    return x > 0.f ? x : SLOPE * x;
}

// ----------------------------------------------------- segment max / sum ----
__global__ void edge_logit_max(const int* __restrict__ src, const int* __restrict__ dst,
                               const float* __restrict__ as, const float* __restrict__ ad,
                               float* __restrict__ m, long long E, int H) {
    long long idx = (long long)blockIdx.x * blockDim.x + threadIdx.x;
    if (idx >= E * H) return;
    const long long e = idx / H;
    const int head = (int)(idx - e * H);
    const float l = leaky(as[(size_t)src[e] * H + head] + ad[(size_t)dst[e] * H + head]);
    atomicMaxF(&m[(size_t)dst[e] * H + head], l);
}

__global__ void edge_expsum(const int* __restrict__ src, const int* __restrict__ dst,
                            const float* __restrict__ as, const float* __restrict__ ad,
                            const float* __restrict__ m, float* __restrict__ denom,
                            long long E, int H) {
    long long idx = (long long)blockIdx.x * blockDim.x + threadIdx.x;
    if (idx >= E * H) return;
    const long long e = idx / H;
    const int head = (int)(idx - e * H);
    const size_t dh = (size_t)dst[e] * H + head;
    const float l = leaky(as[(size_t)src[e] * H + head] + ad[dh]);
    atomicAdd(&denom[dh], __expf(l - m[dh]));
}

// ------------------------------------------------ attention-weighted scatter -
// One block per edge; per-head softmax coefficients staged in LDS once, then
// a pure gather -> fma -> global_atomic_add_f32 channel stream.
__global__ void edge_aggregate(const int* __restrict__ src, const int* __restrict__ dst,
                               const float* __restrict__ h,
                               const float* __restrict__ as, const float* __restrict__ ad,
                               const float* __restrict__ m, const float* __restrict__ denom,
                               float* __restrict__ out, int H, int C) {
    __shared__ float scoef[HEADS];
    const long long e = blockIdx.x;
    const int s = src[e], d = dst[e];
    if (threadIdx.x < (unsigned)H) {
        const int head = threadIdx.x;
        const size_t dh = (size_t)d * H + head;
        const float l = leaky(as[(size_t)s * H + head] + ad[dh]);
        scoef[head] = __expf(l - m[dh]) / (denom[dh] + 1e-16f);
    }
    __syncthreads();
    const int HC = H * C;
    const float* __restrict__ hs = h + (size_t)s * HC;
    float* __restrict__ od = out + (size_t)d * HC;
    for (int i = threadIdx.x; i < HC; i += blockDim.x)
        atomicAdd(&od[i], scoef[i / C] * hs[i]);
}

// ------------------------------------------------------------- bias + ELU ---
__global__ void bias_elu(float* __restrict__ y, const float* __restrict__ b,
                         int cols, size_t n) {
    size_t i = (size_t)blockIdx.x * blockDim.x + threadIdx.x;
    if (i >= n) return;
    const float v = y[i] + b[i % cols];
    y[i] = v > 0.f ? v : __expf(v) - 1.f;
}

// ---------------------------------------------------------------------------
extern "C" void kernel_launch(void* const* d_in, const int* in_sizes, int n_in,
                              void* d_out, int out_size, void* d_ws, size_t ws_size,
                              hipStream_t stream) {
    const float* x      = (const float*)d_in[0];
    const int*   eidx   = (const int*)d_in[1];
    const float* W1     = (const float*)d_in[2];
    const float* a1_src = (const float*)d_in[3];
    const float* a1_dst = (const float*)d_in[4];
    const float* b1     = (const float*)d_in[5];
    const float* W2     = (const float*)d_in[6];
    const float* a2_src = (const float*)d_in[7];
    const float* a2_dst = (const float*)d_in[8];
    const float* b2     = (const float*)d_in[9];

    const int N = in_sizes[0] / FIN;               // 50000
    const long long E = (long long)in_sizes[1] / 2;
    const int* src = eidx;
    const int* dst = eidx + E;

    // workspace carve-up (floats), ~225 MB total
    float* h1   = (float*)d_ws;                    // N*512
    float* out1 = h1   + (size_t)N * HEADS * HID;  // N*512
    float* as1  = out1 + (size_t)N * HEADS * HID;  // N*8
    float* ad1  = as1  + (size_t)N * HEADS;
    float* m1   = ad1  + (size_t)N * HEADS;
    float* dn1  = m1   + (size_t)N * HEADS;
    float* h2   = dn1  + (size_t)N * HEADS;        // N*64
    float* as2  = h2   + (size_t)N * HID;          // N
    float* ad2  = as2  + (size_t)N;
    float* m2   = ad2  + (size_t)N;
    float* dn2  = m2   + (size_t)N;
    float* out2 = (float*)d_out;                   // N*64

    const int T = 256;
    auto blks = [](size_t n, int t) { return (unsigned)((n + t - 1) / t); };

    // ---- layer 1 ----
    const size_t nh1 = (size_t)N * HEADS * HID;
    fill_f32<<<blks((size_t)N * HEADS, T), T, 0, stream>>>(m1, -INFINITY, (size_t)N * HEADS);
    fill_f32<<<blks((size_t)N * HEADS, T), T, 0, stream>>>(dn1, 0.f, (size_t)N * HEADS);
    fill_f32<<<blks(nh1, T), T, 0, stream>>>(out1, 0.f, nh1);

    {   // h1 = x @ W1 : M=N, K=128, Nc=512 -> 3125*8 strips, 8 waves/block
        const int strips = (N / 16) * ((HEADS * HID) / 64);
        wmma_gemm_f32<<<(strips + 7) / 8, 256, 0, stream>>>(x, W1, h1, N, FIN, HEADS * HID);
    }
    node_alpha<<<N, 32 * HEADS, 0, stream>>>(h1, a1_src, a1_dst, as1, ad1, HEADS, HID);
    edge_logit_max<<<blks((size_t)E * HEADS, T), T, 0, stream>>>(src, dst, as1, ad1, m1, E, HEADS);
    edge_expsum<<<blks((size_t)E * HEADS, T), T, 0, stream>>>(src, dst, as1, ad1, m1, dn1, E, HEADS);
    edge_aggregate<<<(unsigned)E, 256, 0, stream>>>(src, dst, h1, as1, ad1, m1, dn1, out1, HEADS, HID);
    bias_elu<<<blks(nh1, T), T, 0, stream>>>(out1, b1, HEADS * HID, nh1);

    // ---- layer 2 ----
    const size_t nh2 = (size_t)N * HID;
    fill_f32<<<blks((size_t)N, T), T, 0, stream>>>(m2, -INFINITY, (size_t)N);
    fill_f32<<<blks((size_t)N, T), T, 0, stream>>>(dn2, 0.f, (size_t)N);
    fill_f32<<<blks(nh2, T), T, 0, stream>>>(out2, 0.f, nh2);

    {   // h2 = elu(out1) @ W2 : M=N, K=512, Nc=64 -> 3125 strips
        const int strips = (N / 16) * (HID / 64);
        wmma_gemm_f32<<<(strips + 7) / 8, 256, 0, stream>>>(out1, W2, h2, N, HEADS * HID, HID);
    }
    node_alpha<<<N, 32, 0, stream>>>(h2, a2_src, a2_dst, as2, ad2, 1, HID);
    edge_logit_max<<<blks((size_t)E, T), T, 0, stream>>>(src, dst, as2, ad2, m2, E, 1);
    edge_expsum<<<blks((size_t)E, T), T, 0, stream>>>(src, dst, as2, ad2, m2, dn2, E, 1);
    edge_aggregate<<<(unsigned)E, 64, 0, stream>>>(src, dst, h2, as2, ad2, m2, dn2, out2, 1, HID);
    bias_elu<<<blks(nh2, T), T, 0, stream>>>(out2, b2, HID, nh2);
}